// GCNBaseline_46901042872928
// MI455X (gfx1250) — compile-verified
//
#include <hip/hip_runtime.h>
#include <hip/hip_bf16.h>

// ---------------------------------------------------------------------------
// GCN forward on MI455X (gfx1250, wave32).
//   h1  = X @ W1                  (WMMA f32 16x16x4, A+B staged in LDS)
//   a1  = scatter_add(norm * h1)  (f32 atomics, L2-resident)
//   h2  = relu(a1 + b1) @ W2      (bias+relu fused into LDS staging)
//   a2  = scatter_add(norm * h2)
//   out = (a2 + b2) @ Wc + bc
// W is staged global->LDS with GLOBAL_LOAD_ASYNC_TO_LDS_B128 when the
// toolchain exposes the builtin (ASYNCcnt / s_wait_asynccnt path).
// ---------------------------------------------------------------------------

typedef __attribute__((ext_vector_type(2))) float v2f;
typedef __attribute__((ext_vector_type(8))) float v8f;
typedef int v4i_ __attribute__((vector_size(16)));   // matches builtin proto

#define N_NODES 50000
#define N_EDGES 800000
#define DIM 128

#if __has_builtin(__builtin_amdgcn_global_load_async_to_lds_b128)
#define HAVE_ASYNC_LDS 1
#else
#define HAVE_ASYNC_LDS 0
#endif

__device__ __forceinline__ void wait_async_zero() {
#if __has_builtin(__builtin_amdgcn_s_wait_asynccnt)
    __builtin_amdgcn_s_wait_asynccnt(0);
#else
    asm volatile("s_wait_asynccnt 0" ::: "memory");
#endif
}

// ---------------- degree / norm ----------------

__global__ void k_init_deg(float* deg, int n) {
    int i = blockIdx.x * blockDim.x + threadIdx.x;
    if (i < n) deg[i] = 1.0f;   // self-loop contributes 1 to every node's degree
}

__global__ void k_count_deg(const int* dst, float* deg, int ne) {
    int e = blockIdx.x * blockDim.x + threadIdx.x;
    if (e < ne) atomicAdd(&deg[dst[e]], 1.0f);
}

__global__ void k_rsqrt(float* deg, int n) {
    int i = blockIdx.x * blockDim.x + threadIdx.x;
    if (i < n) deg[i] = rsqrtf(deg[i]);   // deg >= 1 always (self loop)
}

// ---------------- fused GEMM (16x16 tile / wave, K step 4, fp32 WMMA) ------
// Block = 256 threads = 8 waves; wave w owns output columns [16w, 16w+16).
// Block b owns rows [16b, 16b+16). A tile and the full 128x128 W are staged
// in LDS before the K loop (so the K loop is pure ds_load + v_wmma, and the
// layer-2 in-place agg overwrite stays safe). Epilogue writes h and the
// self-loop-initialized aggregation buffer agg = h * dis^2.

template <int APPLY_BIAS_RELU>
__global__ __launch_bounds__(256)
void k_gemm_fused(const float* __restrict__ X, const float* __restrict__ W,
                  const float* __restrict__ bias_in,
                  const float* __restrict__ dis,
                  float* __restrict__ h_out, float* __restrict__ agg_out,
                  int nrows) {
    __shared__ float As[16][DIM + 4];   // +4 pad: kills 16-way bank conflicts
    __shared__ float Ws[DIM * DIM];     // full 64KB weight tile, row-major

    const int tid = threadIdx.x;
    const int m0  = blockIdx.x * 16;

    // ---- Stage W (128x128 f32) into LDS: async DMA path if available ----
#if HAVE_ASYNC_LDS
    {
        // Cast through integers: AS(1) ptr = 64-bit global VA; AS(3) ptr =
        // low 32 bits of the generic shared pointer (LDS offset on AMDGPU).
        __attribute__((address_space(1))) v4i_* gW =
            (__attribute__((address_space(1))) v4i_*)
                (unsigned long long)(const void*)W;
        __attribute__((address_space(3))) v4i_* lW =
            (__attribute__((address_space(3))) v4i_*)
                (unsigned int)(unsigned long long)(void*)&Ws[0];
        for (int idx = tid; idx < DIM * DIM / 4; idx += 256) {
            __builtin_amdgcn_global_load_async_to_lds_b128(
                gW + idx, lW + idx, 0, 0);
        }
        wait_async_zero();   // drain this wave's ASYNCcnt before the barrier
    }
#else
    for (int idx = tid; idx < DIM * DIM / 4; idx += 256)
        ((float4*)Ws)[idx] = ((const float4*)W)[idx];
#endif

    // ---- Stage A tile (16x128), optionally fused relu(x + b), b128 loads --
    for (int idx = tid; idx < 16 * DIM / 4; idx += 256) {
        const int r  = idx >> 5;        // row in tile
        const int c4 = idx & 31;        // float4 column
        const int row = m0 + r;
        float4 v = make_float4(0.f, 0.f, 0.f, 0.f);
        if (row < nrows) {
            v = ((const float4*)(X + (long)row * DIM))[c4];
            if (APPLY_BIAS_RELU) {
                const float4 b = ((const float4*)bias_in)[c4];
                v.x = fmaxf(v.x + b.x, 0.f);
                v.y = fmaxf(v.y + b.y, 0.f);
                v.z = fmaxf(v.z + b.z, 0.f);
                v.w = fmaxf(v.w + b.w, 0.f);
            }
        }
        *(float4*)&As[r][c4 * 4] = v;   // row stride 132 floats, 16B aligned
    }
    __syncthreads();

    const int wave = tid >> 5;          // 0..7 -> N-tile
    const int lane = tid & 31;
    const int half = lane >> 4;         // 0: lanes 0-15, 1: lanes 16-31
    const int lm   = lane & 15;
    const int n0   = wave * 16;

    v8f acc = {};
    // K loop: 32 x V_WMMA_F32_16X16X4_F32, all operands from LDS.
    for (int k = 0; k < DIM; k += 4) {
        v2f a, b;
        // A 16x4 layout: lanes 0-15 hold {K=k,k+1}, lanes 16-31 hold {K=k+2,k+3}
        a.x = As[lm][k + 2 * half];
        a.y = As[lm][k + 2 * half + 1];
        // B 4x16 layout mirrors A: VGPR0 rows {k, k+2}, VGPR1 rows {k+1, k+3}
        b.x = Ws[(k + 2 * half) * DIM + n0 + lm];
        b.y = Ws[(k + 2 * half + 1) * DIM + n0 + lm];
        acc = __builtin_amdgcn_wmma_f32_16x16x4_f32(
            /*neg_a=*/false, a, /*neg_b=*/false, b,
            /*c_mod=*/(short)0, acc, /*reuse_a=*/false, /*reuse_b=*/false);
    }

    // Epilogue: C VGPR v -> row m0 + v + 8*half, col n0 + lm.
    const int col = n0 + lm;
#pragma unroll
    for (int v = 0; v < 8; ++v) {
        const int row = m0 + v + 8 * half;
        if (row < nrows) {
            const float cv = acc[v];
            const float d  = dis[row];
            h_out[(long)row * DIM + col]   = cv;
            agg_out[(long)row * DIM + col] = cv * d * d;   // self-loop term
        }
    }
}

// ---------------- edge scatter: one wave per edge --------------------------

__global__ __launch_bounds__(256)
void k_scatter(const int* __restrict__ src, const int* __restrict__ dst,
               const float* __restrict__ dis,
               const float* __restrict__ h, float* __restrict__ agg, int ne) {
    const int gtid = blockIdx.x * blockDim.x + threadIdx.x;
    const int e    = gtid >> 5;
    const int lane = gtid & 31;
    if (e >= ne) return;

    const int s = src[e];
    const int d = dst[e];
    const float nrm = dis[s] * dis[d];

    const float4 v = ((const float4*)(h + (long)s * DIM))[lane];
    float* ap = agg + (long)d * DIM + lane * 4;
    atomicAdd(ap + 0, v.x * nrm);
    atomicAdd(ap + 1, v.y * nrm);
    atomicAdd(ap + 2, v.z * nrm);
    atomicAdd(ap + 3, v.w * nrm);
}

// ---------------- classifier head: (a2 + b2) @ Wc + bc ---------------------

__global__ __launch_bounds__(256)
void k_classify(const float* __restrict__ agg, const float* __restrict__ b2,
                const float* __restrict__ Wc, const float* __restrict__ bc,
                float* __restrict__ out, int n) {
    const int i = blockIdx.x * blockDim.x + threadIdx.x;
    if (i >= n) return;
    float a0 = bc[0], a1 = bc[1];
    const float4* row4 = (const float4*)(agg + (long)i * DIM);
    const float4* b4   = (const float4*)b2;
    const float4* w4   = (const float4*)Wc;   // [(d,0),(d,1),(d+1,0),(d+1,1)]
#pragma unroll 4
    for (int q = 0; q < DIM / 4; ++q) {
        const float4 v  = row4[q];
        const float4 bb = b4[q];
        const float4 w01 = w4[q * 2 + 0];   // rows 4q, 4q+1
        const float4 w23 = w4[q * 2 + 1];   // rows 4q+2, 4q+3
        const float e0 = v.x + bb.x, e1 = v.y + bb.y;
        const float e2 = v.z + bb.z, e3 = v.w + bb.w;
        a0 += e0 * w01.x + e1 * w01.z + e2 * w23.x + e3 * w23.z;
        a1 += e0 * w01.y + e1 * w01.w + e2 * w23.y + e3 * w23.w;
    }
    out[i * 2 + 0] = a0;
    out[i * 2 + 1] = a1;
}

// ---------------- host-side launcher ---------------------------------------

extern "C" void kernel_launch(void* const* d_in, const int* in_sizes, int n_in,
                              void* d_out, int out_size, void* d_ws, size_t ws_size,
                              hipStream_t stream) {
    (void)in_sizes; (void)n_in; (void)out_size; (void)ws_size;

    const float* x   = (const float*)d_in[0];
    const int*   ei  = (const int*)  d_in[1];   // [2, E]: src row then dst row
    const float* W1  = (const float*)d_in[2];
    const float* b1  = (const float*)d_in[3];
    const float* W2  = (const float*)d_in[4];
    const float* b2  = (const float*)d_in[5];
    const float* Wc  = (const float*)d_in[6];
    const float* bc  = (const float*)d_in[7];
    float*       out = (float*)d_out;

    const int* src = ei;
    const int* dst = ei + N_EDGES;

    // Workspace layout (floats): dis | bufA (h) | bufB (agg)
    float* dis  = (float*)d_ws;
    float* bufA = dis + ((N_NODES + 63) & ~63);           // h
    float* bufB = bufA + (long)N_NODES * DIM;             // agg

    const int TB = 256;
    const int nodeBlocks = (N_NODES + TB - 1) / TB;
    const int edgeBlocks = (N_EDGES + TB - 1) / TB;
    const int gemmBlocks = (N_NODES + 15) / 16;
    const int scatBlocks = (N_EDGES * 32 + TB - 1) / TB;  // one wave per edge

    // norm = dis[src] * dis[dst], dis = deg^-1/2 with self loops
    k_init_deg<<<nodeBlocks, TB, 0, stream>>>(dis, N_NODES);
    k_count_deg<<<edgeBlocks, TB, 0, stream>>>(dst, dis, N_EDGES);
    k_rsqrt<<<nodeBlocks, TB, 0, stream>>>(dis, N_NODES);

    // Layer 1: h1 = X@W1 ; agg1 = self-loop init, then edge scatter.
    k_gemm_fused<0><<<gemmBlocks, TB, 0, stream>>>(x, W1, nullptr, dis,
                                                   bufA, bufB, N_NODES);
    k_scatter<<<scatBlocks, TB, 0, stream>>>(src, dst, dis, bufA, bufB, N_EDGES);

    // Layer 2: h2 = relu(agg1+b1)@W2 (staged through LDS -> in-place agg safe).
    k_gemm_fused<1><<<gemmBlocks, TB, 0, stream>>>(bufB, W2, b1, dis,
                                                   bufA, bufB, N_NODES);
    k_scatter<<<scatBlocks, TB, 0, stream>>>(src, dst, dis, bufA, bufB, N_EDGES);

    // Head: logits = (agg2 + b2) @ Wc + bc
    k_classify<<<nodeBlocks, TB, 0, stream>>>(bufB, b2, Wc, bc, out, N_NODES);
}